// AttentionGCNConv_40578851012971
// MI455X (gfx1250) — compile-verified
//
#include <hip/hip_runtime.h>
#include <hip/hip_bf16.h>

typedef __attribute__((ext_vector_type(2))) float v2f;
typedef __attribute__((ext_vector_type(4))) float v4f;
typedef __attribute__((ext_vector_type(8))) float v8f;

// ---------------------------------------------------------------------------
// Kernel 1: h = x @ W_lin + b_lin  via native fp32 WMMA (V_WMMA_F32_16X16X4_F32)
// One wave per 16-row tile, grid-striding over tiles with the 16 B-fragments
// (whole 64x16 W) held in registers for the wave's lifetime.
// A 16x4 layout (ISA 7.12.2): lanes 0-15 hold M=lane, {K=k0, K=k0+1};
//                             lanes 16-31 hold M=lane-16, {K=k0+2, K=k0+3}.
// B 4x16 symmetric. C/D: VGPR r -> (M=r, N=lane) / (M=r+8, N=lane-16).
// ---------------------------------------------------------------------------
__global__ void node_gemm_wmma(const float* __restrict__ x,     // [N,64]
                               const float* __restrict__ W,     // [64,16]
                               const float* __restrict__ bias,  // [16]
                               float* __restrict__ h,           // [N,16]
                               int nTiles) {
  const int lane   = threadIdx.x & 31;
  const int wave   = blockIdx.x * (blockDim.x >> 5) + (threadIdx.x >> 5);
  const int nWaves = gridDim.x * (blockDim.x >> 5);
  const int half   = lane >> 4;        // 0: lanes 0-15, 1: lanes 16-31
  const int lm     = lane & 15;

  // Preload all 16 B fragments (entire W) once per wave: 32 VGPRs.
  v2f Breg[16];
#pragma unroll
  for (int s = 0; s < 16; ++s) {
    const int k0 = 4 * s + 2 * half;
    Breg[s].x = W[(size_t)k0 * 16 + lm];
    Breg[s].y = W[(size_t)(k0 + 1) * 16 + lm];
  }
  const float bv = bias[lm];

  for (int t = wave; t < nTiles; t += nWaves) {   // wave-uniform trip count
    const int row0 = t * 16;
    const float* xr = x + (size_t)(row0 + lm) * 64 + 2 * half;
    v8f acc = {};
#pragma unroll
    for (int s = 0; s < 16; ++s) {
      // A pair (K=k0,k0+1): 8B-aligned contiguous -> global_load_b64
      v2f a = *(const v2f*)(xr + 4 * s);
      acc = __builtin_amdgcn_wmma_f32_16x16x4_f32(
          false, a, false, Breg[s], (short)0, acc, false, false);
    }
#pragma unroll
    for (int r = 0; r < 8; ++r) {
      const int row = row0 + r + 8 * half;
      h[(size_t)row * 16 + lm] = acc[r] + bv;
    }
  }
}

// ---------------------------------------------------------------------------
// Kernel 2: one edge per wave32, one of the 32 "combined" scalars per lane.
//   lane 0..15  : c = nv[lane] = h[col[e]*16 + lane]
//   lane 16..31 : c = ea[lane-16] = edge_attr[e,:] . W_edge[:,lane-16] + b_edge
// Per-scalar MLP vectorized as float2 (-> v_pk_fma_f32), softmax across the
// wave with shfl_xor, repeat_interleave via shfl from lane>>1.
// ---------------------------------------------------------------------------
__global__ void edge_attn_kernel(const float* __restrict__ h,          // [N,16]
                                 const float* __restrict__ edge_attr,  // [E,8]
                                 const int*   __restrict__ col,        // [E]
                                 const float* __restrict__ W_edge,     // [8,16]
                                 const float* __restrict__ b_edge,     // [16]
                                 const float* __restrict__ w1,         // [16]
                                 const float* __restrict__ b1,         // [16]
                                 const float* __restrict__ w2,         // [16]
                                 const float* __restrict__ b2p,        // [1]
                                 float* __restrict__ out,              // [E,32]
                                 int E) {
  const int lane    = threadIdx.x & 31;
  const int waveId  = blockIdx.x * (blockDim.x >> 5) + (threadIdx.x >> 5);
  const int nWaves  = gridDim.x * (blockDim.x >> 5);

  // Preload MLP params as float2 pairs (amortized over the grid-stride loop).
  const v2f* w1v = (const v2f*)w1;
  const v2f* b1v = (const v2f*)b1;
  const v2f* w2v = (const v2f*)w2;
  v2f w1p[8], b1p[8], w2p[8];
#pragma unroll
  for (int j = 0; j < 8; ++j) { w1p[j] = w1v[j]; b1p[j] = b1v[j]; w2p[j] = w2v[j]; }
  const float b2 = b2p[0];

  // Per-lane W_edge column (used by lanes 16..31; harmless for others).
  const int ec = lane & 15;
  float wec[8];
#pragma unroll
  for (int d = 0; d < 8; ++d) wec[d] = W_edge[(size_t)d * 16 + ec];
  const float bec = b_edge[ec];

  for (int e = waveId; e < E; e += nWaves) {
    const int src = col[e];
    float c;
    if (lane < 16) {
      c = h[(size_t)src * 16 + lane];                 // nv[lane]
    } else {
      // edge_attr row: 32B, 32B-aligned -> two global_load_b128
      const v4f* eap = (const v4f*)(edge_attr + (size_t)e * 8);
      const v4f e0 = eap[0];
      const v4f e1 = eap[1];
      float acc = bec;
      acc = fmaf(e0.x, wec[0], acc);
      acc = fmaf(e0.y, wec[1], acc);
      acc = fmaf(e0.z, wec[2], acc);
      acc = fmaf(e0.w, wec[3], acc);
      acc = fmaf(e1.x, wec[4], acc);
      acc = fmaf(e1.y, wec[5], acc);
      acc = fmaf(e1.z, wec[6], acc);
      acc = fmaf(e1.w, wec[7], acc);
      c = acc;                                         // ea[lane-16]
    }

    // score = b2 + sum_j w2[j] * relu(c*w1[j] + b1[j])  (packed f32 pairs)
    v2f c2;  c2.x = c;  c2.y = c;
    v2f acc2 = {0.f, 0.f};
    const v2f zero2 = {0.f, 0.f};
#pragma unroll
    for (int j = 0; j < 8; ++j) {
      v2f t = c2 * w1p[j] + b1p[j];
      t = __builtin_elementwise_max(t, zero2);
      acc2 = acc2 + t * w2p[j];
    }
    const float s = acc2.x + acc2.y + b2;

    // softmax over all 32 lanes
    float m = s;
#pragma unroll
    for (int off = 16; off > 0; off >>= 1)
      m = fmaxf(m, __shfl_xor(m, off, 32));
    const float ex = __expf(s - m);
    float sum = ex;
#pragma unroll
    for (int off = 16; off > 0; off >>= 1)
      sum += __shfl_xor(sum, off, 32);
    const float attn = ex * __builtin_amdgcn_rcpf(sum);  // v_rcp_f32

    // out[e, lane] = nv[lane/2] * attn ; nv[k] lives in lane k (k<16)
    const float nvv = __shfl(c, lane >> 1, 32);
    out[(size_t)e * 32 + lane] = nvv * attn;
  }
}

extern "C" void kernel_launch(void* const* d_in, const int* in_sizes, int n_in,
                              void* d_out, int out_size, void* d_ws, size_t ws_size,
                              hipStream_t stream) {
  const float* x         = (const float*)d_in[0];
  const float* edge_attr = (const float*)d_in[1];
  const int*   col       = (const int*)  d_in[2];
  const float* W_lin     = (const float*)d_in[3];
  const float* b_lin     = (const float*)d_in[4];
  const float* W_edge    = (const float*)d_in[5];
  const float* b_edge    = (const float*)d_in[6];
  const float* w1        = (const float*)d_in[7];
  const float* b1        = (const float*)d_in[8];
  const float* w2        = (const float*)d_in[9];
  const float* b2        = (const float*)d_in[10];
  float* out = (float*)d_out;

  const int N = in_sizes[0] / 64;   // IN_C = 64
  const int E = in_sizes[2];

  float* h = (float*)d_ws;          // [N,16] fp32 = N*64 bytes

  // Node GEMM: waves grid-stride over 16-row tiles with W held in registers.
  const int nTiles = (N + 15) / 16;                 // 6250 for N=100000
  const int gemmBlocks = 256;                        // 2048 waves, ~3 tiles each
  node_gemm_wmma<<<gemmBlocks, 256, 0, stream>>>(x, W_lin, b_lin, h, nTiles);

  // Edge kernel: grid-stride over edges, one edge per wave.
  const int blocks = 2048;          // 16384 waves; each handles ~E/16384 edges
  edge_attn_kernel<<<blocks, 256, 0, stream>>>(h, edge_attr, col, W_edge, b_edge,
                                               w1, b1, w2, b2, out, E);
}